// VolGeoNet_38500086841605
// MI455X (gfx1250) — compile-verified
//
#include <hip/hip_runtime.h>
#include <hip/hip_bf16.h>

// Trilinear interpolation (value + 256-ch feature) on a 65^3 grid — MI455X.
// Bandwidth-bound gather (~2.15 GB issued feature reads, ~270 MB writes vs
// 23.3 TB/s). Design:
//   - 1 block = 16 points, 128 threads (4 wave32).
//   - feat(16x256) = W(16x128) x F_g(128x256) via V_WMMA_F32_16X16X4_F32.
//   - Gather: CDNA5 async global->LDS b128 (ASYNCcnt), 256B granules,
//     register-resident row pointers (no per-load LDS index fetch).
//   - Double-buffered LDS: chunk nc+1's gathers are issued before computing
//     chunk nc, so WMMA + ds traffic hides HBM gather latency.
//   - LDS row stride 72 dwords -> the two lane-halves of a B-fragment load
//     (rows K, K+2) hit disjoint 16-bank ranges.

#define NG           64
#define NP1          65
#define WFEAT        256
#define PTS_PER_BLK  16
#define NTHREADS     128
#define CHUNK        64                 // channels staged per pass
#define LSTRIDE      72                 // dwords; 2*72 % 64 == 16
#define K_TOT        128                // 16 points * 8 corners
#define HALF_DW      (K_TOT * LSTRIDE)  // dwords per LDS buffer half

typedef __attribute__((ext_vector_type(2))) float v2f;
typedef __attribute__((ext_vector_type(8))) float v8f;

#define USE_ASYNC_GATHER 1

__global__ __launch_bounds__(NTHREADS) void trilerp_wmma_kernel(
    const float* __restrict__ x,          // (B,3)
    const float* __restrict__ gval,       // (65^3, 1)
    const float* __restrict__ gfeat,      // (65^3, 256)
    float* __restrict__ out_val,          // (B,)
    float* __restrict__ out_feat)         // (B,256)
{
    __shared__ float wA[K_TOT];               // weights, k = p*8 + corner
    __shared__ int   flats[K_TOT];            // gathered row indices
    __shared__ float fbuf[2 * HALF_DW];       // 72 KB double-buffered rows

    const int t  = threadIdx.x;
    const int pl = t >> 3;                    // local point 0..15
    const int c  = t & 7;                     // corner 0..7
    const int p  = blockIdx.x * PTS_PER_BLK + pl;

    // ---- weights & flat indices for this (point, corner) ----
    const float x0 = x[3 * p + 0];
    const float x1 = x[3 * p + 1];
    const float x2 = x[3 * p + 2];
    const float r0 = (x0 + 1.0f) * 32.0f;     // (x - min_c) * N/SIDE
    const float r1 = (x1 + 1.0f) * 32.0f;
    const float r2 = (x2 + 1.0f) * 32.0f;
    const bool valid = (r0 >= 0.0f) && (r0 <= 64.0f) &&
                       (r1 >= 0.0f) && (r1 <= 64.0f) &&
                       (r2 >= 0.0f) && (r2 <= 64.0f);
    const int i0 = min(max((int)floorf(r0), 0), NG - 1);
    const int i1 = min(max((int)floorf(r1), 0), NG - 1);
    const int i2 = min(max((int)floorf(r2), 0), NG - 1);
    const float t0 = r0 - (float)i0;
    const float t1 = r1 - (float)i1;
    const float t2 = r2 - (float)i2;
    const int ox = (c >> 2) & 1, oy = (c >> 1) & 1, oz = c & 1;
    const float wx = ox ? t0 : 1.0f - t0;
    const float wy = oy ? t1 : 1.0f - t1;
    const float wz = oz ? t2 : 1.0f - t2;
    const float w  = valid ? (wx * wy * wz) : 0.0f;   // w==0 -> outputs 0
    const int flat = ((i0 + ox) * NP1 + (i1 + oy)) * NP1 + (i2 + oz);

    wA[t]    = w;
    flats[t] = flat;

    // ---- value channel: gather + 8-lane reduction ----
    float v = w * gval[flat];
    v += __shfl_xor(v, 1, 32);
    v += __shfl_xor(v, 2, 32);
    v += __shfl_xor(v, 4, 32);
    if (c == 0) out_val[p] = v;

    __syncthreads();

    // ---- per-lane constant A-fragment weights ----
    // A (16x4 f32): lane L -> M = L%16; VGPR r holds K = 4s + r + 2h.
    // W[M][K] nonzero only for K in [8M, 8M+8) -> hits at s = 2m, 2m+1.
    const int lane = t & 31;
    const int wave = t >> 5;
    const int h    = lane >> 4;
    const int m    = lane & 15;
    const float wv0 = wA[m * 8 + 2 * h];
    const float wv1 = wA[m * 8 + 2 * h + 1];
    const float wv2 = wA[m * 8 + 4 + 2 * h];
    const float wv3 = wA[m * 8 + 4 + 2 * h + 1];

    // ---- register-resident gather row pointers ----
    // Thread t stages row r = it*8 + rowsel (it = 0..15): corner `rowsel&7`
    // of point `it`; 16 lanes (rq) cover one 256B row-chunk contiguously.
    const int rq     = t & 15;
    const int rowsel = t >> 4;
    const float* rowp[16];
    #pragma unroll
    for (int it = 0; it < 16; ++it) {
        rowp[it] = gfeat + (size_t)flats[it * 8 + rowsel] * WFEAT + rq * 4;
    }
    // LDS byte offset of this thread's slot in row (it*8+rowsel) of buffer 0.
    const unsigned ldsbase = (unsigned)(uintptr_t)
        (&fbuf[(size_t)rowsel * LSTRIDE + rq * 4]);

    // ---- staging: 16 async b128 gathers for chunk nc into buffer (nc&1) ----
    auto stage = [&](int nc) {
        const unsigned hoff = (unsigned)(nc & 1) * (HALF_DW * 4u);
        #pragma unroll
        for (int it = 0; it < 16; ++it) {
            const float*   src    = rowp[it] + nc * CHUNK;
            const unsigned ldsoff = ldsbase + hoff + (unsigned)(it * 8 * LSTRIDE * 4);
#if USE_ASYNC_GATHER
            asm volatile("global_load_async_to_lds_b128 %0, %1, off"
                         :: "v"(ldsoff), "v"(src) : "memory");
#else
            const float4 tmp = *(const float4*)src;
            *(float4*)(uintptr_t)(
                (uintptr_t)&fbuf[0] + (ldsoff - (unsigned)(uintptr_t)&fbuf[0])) = tmp;
#endif
        }
    };

    stage(0);

    const int colbase = wave * 16 + m;
    for (int nc = 0; nc < 4; ++nc) {
#if USE_ASYNC_GATHER
#if defined(__has_builtin) && __has_builtin(__builtin_amdgcn_s_wait_asynccnt)
        __builtin_amdgcn_s_wait_asynccnt(0);
#else
        asm volatile("s_wait_asynccnt 0" ::: "memory");
#endif
#endif
        __syncthreads();                 // chunk nc's rows visible to all

        if (nc < 3) stage(nc + 1);       // overlap next gather with compute

        // ---- WMMA: this wave owns channels [nc*64 + wave*16, +16) ----
        const float* fb = &fbuf[(nc & 1) * HALF_DW];
        v8f acc = {0.f, 0.f, 0.f, 0.f, 0.f, 0.f, 0.f, 0.f};
        // depth-2 pipelined B-fragment loads (rows K=k0+2h, k0+1+2h)
        float b0 = fb[(0 + 2 * h) * LSTRIDE + colbase];
        float b1 = fb[(1 + 2 * h) * LSTRIDE + colbase];
        float c0 = fb[(4 + 2 * h) * LSTRIDE + colbase];
        float c1 = fb[(5 + 2 * h) * LSTRIDE + colbase];
        #pragma unroll
        for (int s = 0; s < 32; ++s) {
            float nb0 = 0.0f, nb1 = 0.0f;
            if (s < 30) {
                nb0 = fb[((s + 2) * 4 + 2 * h    ) * LSTRIDE + colbase];
                nb1 = fb[((s + 2) * 4 + 2 * h + 1) * LSTRIDE + colbase];
            }
            const bool hit = ((s >> 1) == m);
            const float a0 = hit ? ((s & 1) ? wv2 : wv0) : 0.0f;
            const float a1 = hit ? ((s & 1) ? wv3 : wv1) : 0.0f;
            v2f A;  A[0] = a0;  A[1] = a1;
            v2f Bf; Bf[0] = b0; Bf[1] = b1;
            acc = __builtin_amdgcn_wmma_f32_16x16x4_f32(
                      false, A, false, Bf, (short)0, acc, false, false);
            b0 = c0; b1 = c1; c0 = nb0; c1 = nb1;
        }

        // ---- store D tile: VGPR r -> row (r + 8h), col = nc*64 + wave*16 + m
        float* dstf = out_feat + (size_t)(blockIdx.x * PTS_PER_BLK) * WFEAT
                      + nc * CHUNK + wave * 16 + m;
        #pragma unroll
        for (int r = 0; r < 8; ++r) {
            dstf[(size_t)(r + 8 * h) * WFEAT] = acc[r];
        }
        __syncthreads();                 // all waves done with buffer (nc&1)
    }
}

extern "C" void kernel_launch(void* const* d_in, const int* in_sizes, int n_in,
                              void* d_out, int out_size, void* d_ws, size_t ws_size,
                              hipStream_t stream) {
    const float* x     = (const float*)d_in[0];
    const float* gval  = (const float*)d_in[1];
    const float* gfeat = (const float*)d_in[2];
    const int B = in_sizes[0] / 3;           // 262144
    float* out_val  = (float*)d_out;         // (B,1) first
    float* out_feat = out_val + B;           // then (B,256)
    dim3 grid(B / PTS_PER_BLK);
    trilerp_wmma_kernel<<<grid, NTHREADS, 0, stream>>>(x, gval, gfeat, out_val, out_feat);
}